// SuperpixelTokenizer_37915971289104
// MI455X (gfx1250) — compile-verified
//
#include <hip/hip_runtime.h>
#include <hip/hip_bf16.h>
#include <math.h>

// Problem constants (fixed by setup_inputs): feat [8,3,512,512] f32
constexpr int Bb = 8;
constexpr int Cc = 3;
constexpr int Hh = 512;
constexpr int Ww = 512;
constexpr int Np = Hh * Ww;          // 262144 pixels per image
constexpr int TILES_X = Ww / 16;     // 32
constexpr int TILES_Y = Hh / 16;     // 32
constexpr float LGRAD = 27.8f;
constexpr float LCOL  = 10.0f;

typedef float v2f __attribute__((ext_vector_type(2)));
typedef float v8f __attribute__((ext_vector_type(8)));

#if defined(__AMDGCN__) && __has_builtin(__builtin_amdgcn_global_load_async_to_lds_b32)
#define HAVE_ASYNC_LDS 1
typedef __attribute__((address_space(1))) int glob_int_t;   // prints as "__device__ int"
typedef __attribute__((address_space(3))) int lds_int_t;    // prints as "__shared__ int"
#else
#define HAVE_ASYNC_LDS 0
#endif

__device__ __forceinline__ void wait_async_zero() {
#if defined(__AMDGCN__)
#if __has_builtin(__builtin_amdgcn_s_wait_asynccnt)
    __builtin_amdgcn_s_wait_asynccnt(0);
#else
    asm volatile("s_wait_asynccnt 0" ::: "memory");
#endif
#endif
}

// ---------------------------------------------------------------------------
// Kernel 1: gradient magnitude via V_WMMA_F32_16X16X4_F32.
// gx = T * Mx  (Mx[k][n] = d(k==n+1) - d(k==n))
// gy = My * T  (My[m][k] = d(k==m+1) - d(k==m))
// One wave32 per 16x16 tile; K=16 done as 4 chained K=4 WMMAs.
// Tiles staged global->LDS with GLOBAL_LOAD_ASYNC_TO_LDS_B32 (ASYNCcnt path)
// when available; VGPR-bounce fallback otherwise.
// A 16x4 f32 layout (ISA 7.12.2): lanes 0-15 rows M, VGPR v -> K = v + 2*half.
// B 4x16 f32 assumed symmetric: lane holds col N=lane%16, VGPR v -> K = v + 2*half.
// C/D 16x16 f32: VGPR v -> M = v + 8*half, N = lane%16.
// ---------------------------------------------------------------------------
__global__ void k_gmag(const float* __restrict__ feat, float* __restrict__ gmag) {
    const int lane = threadIdx.x & 31;
    const int wv   = threadIdx.x >> 5;                 // 8 waves / block
    const int tile = blockIdx.x * 8 + wv;              // 0 .. 8191
    const int b    = tile >> 10;                       // /(32*32)
    const int tr   = (tile >> 5) & 31;                 // tile row
    const int tc   = tile & 31;                        // tile col
    const int half = lane >> 4;
    const int l16  = lane & 15;

    __shared__ float T[8][16][17];                     // +1 pad: bank-conflict-free column reads

    v8f acc = {0.f, 0.f, 0.f, 0.f, 0.f, 0.f, 0.f, 0.f};

    for (int ch = 0; ch < Cc; ++ch) {
        const float* base = feat + ((size_t)(b * Cc + ch)) * Np;

        // stage tile into LDS
#if HAVE_ASYNC_LDS
        #pragma unroll
        for (int i = 0; i < 8; ++i) {
            int idx = lane + 32 * i;
            int r = idx >> 4, w2 = idx & 15;
            __builtin_amdgcn_global_load_async_to_lds_b32(
                (glob_int_t*)&base[(tr * 16 + r) * Ww + tc * 16 + w2],
                (lds_int_t*)&T[wv][r][w2],
                /*offset=*/0, /*cpol=*/0);
        }
        wait_async_zero();
#else
        #pragma unroll
        for (int i = 0; i < 8; ++i) {
            int idx = lane + 32 * i;
            int r = idx >> 4, w2 = idx & 15;
            T[wv][r][w2] = base[(tr * 16 + r) * Ww + tc * 16 + w2];
        }
#endif
        __syncthreads();

        v8f gx = {0.f, 0.f, 0.f, 0.f, 0.f, 0.f, 0.f, 0.f};
        v8f gy = {0.f, 0.f, 0.f, 0.f, 0.f, 0.f, 0.f, 0.f};

        #pragma unroll
        for (int c4 = 0; c4 < 4; ++c4) {
            const int K0 = 4 * c4 + 2 * half;
            // A for gx: rows of T
            v2f a;
            a.x = T[wv][l16][K0];
            a.y = T[wv][l16][K0 + 1];
            // B for gx: Mx column l16 ; identical values serve as A for gy (My row l16)
            v2f m;
            m.x = ((K0     == l16 + 1) ? 1.f : 0.f) - ((K0     == l16) ? 1.f : 0.f);
            m.y = ((K0 + 1 == l16 + 1) ? 1.f : 0.f) - ((K0 + 1 == l16) ? 1.f : 0.f);
            // B for gy: column l16 of T
            v2f bt;
            bt.x = T[wv][K0][l16];
            bt.y = T[wv][K0 + 1][l16];

            gx = __builtin_amdgcn_wmma_f32_16x16x4_f32(false, a, false, m,  (short)0, gx, false, false);
            gy = __builtin_amdgcn_wmma_f32_16x16x4_f32(false, m, false, bt, (short)0, gy, false, false);
        }

        // patch gx column 15 (needs T[r][16] from next tile; zero at image edge)
        if (l16 == 15) {
            const bool edge = (tc == TILES_X - 1);
            #pragma unroll
            for (int v = 0; v < 8; ++v) {
                int mrow = v + 8 * half;
                float t15 = T[wv][mrow][15];
                gx[v] = edge ? 0.f : (base[(tr * 16 + mrow) * Ww + tc * 16 + 16] - t15);
            }
        }
        // patch gy row 15 (VGPR 7, lanes 16-31 hold M=15)
        if (half == 1) {
            const bool edge = (tr == TILES_Y - 1);
            float t15 = T[wv][15][l16];
            gy[7] = edge ? 0.f : (base[(tr * 16 + 16) * Ww + tc * 16 + l16] - t15);
        }

        #pragma unroll
        for (int v = 0; v < 8; ++v)
            acc[v] += sqrtf(gx[v] * gx[v] + gy[v] * gy[v] + 1e-12f);

        __syncthreads();
    }

    float* gout = gmag + (size_t)b * Np;
    #pragma unroll
    for (int v = 0; v < 8; ++v) {
        int mrow = v + 8 * half;
        gout[(tr * 16 + mrow) * Ww + tc * 16 + l16] = acc[v] * (1.f / 3.f);
    }
}

// ---------------------------------------------------------------------------
// label init: labels[b*N + p] = p
// ---------------------------------------------------------------------------
__global__ void k_init(int* __restrict__ labels) {
    int i = blockIdx.x * 256 + threadIdx.x;
    labels[i] = i & (Np - 1);   // Np is a power of two
}

// zero region accumulators for one batch
__global__ void k_zero(float* __restrict__ cnt, float* __restrict__ cs0, float* __restrict__ cs1,
                       float* __restrict__ cs2, float* __restrict__ gsm, float* __restrict__ best,
                       int* __restrict__ bn) {
    int i = blockIdx.x * 256 + threadIdx.x;
    cnt[i] = 0.f; cs0[i] = 0.f; cs1[i] = 0.f; cs2[i] = 0.f;
    gsm[i] = 0.f; best[i] = 0.f; bn[i] = Np;
}

// segment sums (atomics stay in L2; conflict-free at level 0 where labels==id)
__global__ void k_accum(const float* __restrict__ feat_b, const float* __restrict__ gmag_b,
                        const int* __restrict__ labels_b, float* __restrict__ cnt,
                        float* __restrict__ cs0, float* __restrict__ cs1, float* __restrict__ cs2,
                        float* __restrict__ gsm) {
    int p = blockIdx.x * 256 + threadIdx.x;
    __builtin_prefetch(feat_b + p + 2048, 0, 1);           // global_prefetch_b8
    int l = labels_b[p];
    atomicAdd(&cnt[l], 1.f);
    atomicAdd(&cs0[l], feat_b[p]);
    atomicAdd(&cs1[l], feat_b[Np + p]);
    atomicAdd(&cs2[l], feat_b[2 * Np + p]);
    atomicAdd(&gsm[l], gmag_b[p]);
}

// region means, in place
__global__ void k_means(float* __restrict__ cnt, float* __restrict__ cs0, float* __restrict__ cs1,
                        float* __restrict__ cs2, float* __restrict__ gsm) {
    int i = blockIdx.x * 256 + threadIdx.x;
    float s = 1.f / fmaxf(cnt[i], 1.f);
    cs0[i] *= s; cs1[i] *= s; cs2[i] *= s; gsm[i] *= s;
}

__device__ __forceinline__ float sim_pair(const float* cs0, const float* cs1, const float* cs2,
                                          const float* mg, int la, int lb) {
    float d0 = cs0[la] - cs0[lb];
    float d1 = cs1[la] - cs1[lb];
    float d2 = cs2[la] - cs2[lb];
    float dc = sqrtf(d0 * d0 + d1 * d1 + d2 * d2 + 1e-12f);
    float ge = 0.5f * (mg[la] + mg[lb]);
    return expf(-(dc * (1.f / LCOL) + ge * (1.f / LGRAD)));
}

// pass 1: best-neighbor similarity per region (segment_max) — positive floats,
// so int atomicMax on the bit pattern is order-preserving.
__global__ void k_edge_best(const int* __restrict__ labels_b, const float* __restrict__ cs0,
                            const float* __restrict__ cs1, const float* __restrict__ cs2,
                            const float* __restrict__ mg, int* __restrict__ bestbits) {
    int p = blockIdx.x * 256 + threadIdx.x;
    int y = p >> 9, x = p & (Ww - 1);
    int la = labels_b[p];
    int q[4]; bool ok[4];
    ok[0] = (x < Ww - 1); q[0] = p + 1;
    ok[1] = (x > 0);      q[1] = p - 1;
    ok[2] = (y < Hh - 1); q[2] = p + Ww;
    ok[3] = (y > 0);      q[3] = p - Ww;
    #pragma unroll
    for (int k = 0; k < 4; ++k) {
        if (!ok[k]) continue;
        int lb = labels_b[q[k]];
        if (lb == la) continue;
        float s = sim_pair(cs0, cs1, cs2, mg, la, lb);
        atomicMax(&bestbits[la], __float_as_int(s));
    }
}

// pass 2: deterministic best neighbor id (segment_min over argmax candidates)
__global__ void k_edge_bn(const int* __restrict__ labels_b, const float* __restrict__ cs0,
                          const float* __restrict__ cs1, const float* __restrict__ cs2,
                          const float* __restrict__ mg, const int* __restrict__ bestbits,
                          int* __restrict__ bn) {
    int p = blockIdx.x * 256 + threadIdx.x;
    int y = p >> 9, x = p & (Ww - 1);
    int la = labels_b[p];
    int bb = bestbits[la];
    int q[4]; bool ok[4];
    ok[0] = (x < Ww - 1); q[0] = p + 1;
    ok[1] = (x > 0);      q[1] = p - 1;
    ok[2] = (y < Hh - 1); q[2] = p + Ww;
    ok[3] = (y > 0);      q[3] = p - Ww;
    #pragma unroll
    for (int k = 0; k < 4; ++k) {
        if (!ok[k]) continue;
        int lb = labels_b[q[k]];
        if (lb == la) continue;
        float s = sim_pair(cs0, cs1, cs2, mg, la, lb);
        if (__float_as_int(s) == bb) atomicMin(&bn[la], lb);
    }
}

// mutual-best, cycle-free merge decision
__global__ void k_parent(const int* __restrict__ bn, const int* __restrict__ bestbits,
                         int* __restrict__ par) {
    int r = blockIdx.x * 256 + threadIdx.x;
    int bnv  = bn[r];
    int bnc  = min(max(bnv, 0), Np - 1);
    int back = bn[bnc];
    int backc = min(max(back, 0), Np - 1);
    bool mutual = (bnv < Np) && (backc == r) && (__int_as_float(bestbits[r]) > 0.f);
    par[r] = mutual ? min(r, bnc) : r;
}

__global__ void k_relabel(int* __restrict__ labels_b, const int* __restrict__ par) {
    int p = blockIdx.x * 256 + threadIdx.x;
    labels_b[p] = par[labels_b[p]];
}

// scatter region-mean colors back to pixels
__global__ void k_colmap(const int* __restrict__ labels_b, const float* __restrict__ cs0,
                         const float* __restrict__ cs1, const float* __restrict__ cs2,
                         float* __restrict__ out_b) {
    int p = blockIdx.x * 256 + threadIdx.x;
    int l = labels_b[p];
    out_b[p]          = cs0[l];
    out_b[Np + p]     = cs1[l];
    out_b[2 * Np + p] = cs2[l];
}

// ---------------------------------------------------------------------------
extern "C" void kernel_launch(void* const* d_in, const int* in_sizes, int n_in,
                              void* d_out, int out_size, void* d_ws, size_t ws_size,
                              hipStream_t stream) {
    const float* feat = (const float*)d_in[0];
    int*   labels = (int*)d_out;                       // [B,N] int32
    float* colmap = (float*)d_out + (size_t)Bb * Np;   // [B,3,N] f32

    // workspace layout (16 MB): gmag[B*N] + per-batch region arrays
    float* gmag = (float*)d_ws;
    float* cnt  = gmag + (size_t)Bb * Np;
    float* cs0  = cnt + Np;
    float* cs1  = cs0 + Np;
    float* cs2  = cs1 + Np;
    float* gsm  = cs2 + Np;
    float* best = gsm + Np;
    int*   bn   = (int*)(best + Np);
    int*   par  = bn + Np;

    const int PB = Np / 256;          // 1024 blocks of 256 for per-pixel kernels

    k_gmag<<<(Bb * TILES_X * TILES_Y) / 8, 256, 0, stream>>>(feat, gmag);
    k_init<<<(Bb * Np) / 256, 256, 0, stream>>>(labels);

    for (int b = 0; b < Bb; ++b) {
        const float* fb = feat + (size_t)b * Cc * Np;
        const float* gb = gmag + (size_t)b * Np;
        int* lb = labels + (size_t)b * Np;

        for (int lvl = 0; lvl < 4; ++lvl) {
            k_zero     <<<PB, 256, 0, stream>>>(cnt, cs0, cs1, cs2, gsm, best, bn);
            k_accum    <<<PB, 256, 0, stream>>>(fb, gb, lb, cnt, cs0, cs1, cs2, gsm);
            k_means    <<<PB, 256, 0, stream>>>(cnt, cs0, cs1, cs2, gsm);
            k_edge_best<<<PB, 256, 0, stream>>>(lb, cs0, cs1, cs2, gsm, (int*)best);
            k_edge_bn  <<<PB, 256, 0, stream>>>(lb, cs0, cs1, cs2, gsm, (int*)best, bn);
            k_parent   <<<PB, 256, 0, stream>>>(bn, (int*)best, par);
            k_relabel  <<<PB, 256, 0, stream>>>(lb, par);
        }
        // final region means -> color map
        k_zero  <<<PB, 256, 0, stream>>>(cnt, cs0, cs1, cs2, gsm, best, bn);
        k_accum <<<PB, 256, 0, stream>>>(fb, gb, lb, cnt, cs0, cs1, cs2, gsm);
        k_means <<<PB, 256, 0, stream>>>(cnt, cs0, cs1, cs2, gsm);
        k_colmap<<<PB, 256, 0, stream>>>(lb, cs0, cs1, cs2, colmap + (size_t)b * Cc * Np);
    }
}